// Net_62191126446411
// MI455X (gfx1250) — compile-verified
//
#include <hip/hip_runtime.h>
#include <hip/hip_bf16.h>
#include <math.h>

// ---------------------------------------------------------------------------
// AdderNet fused forward for MI455X (gfx1250, wave32).
//   out[b,o,l] = -sum_p |patch[b,p,l] - w[o,p]|   (L1 distance -> VALU, not WMMA)
// One block per image; all intermediates live in LDS; CDNA5 async global->LDS
// staging (ASYNCcnt path); weights repacked transposed into LDS so hot-loop
// weight fetches are single ds_load_b128/b64 ops instead of global loads.
// ---------------------------------------------------------------------------

#ifndef __has_builtin
#define __has_builtin(x) 0
#endif

#if __has_builtin(__builtin_amdgcn_global_load_async_to_lds_b128)
#define HAVE_ASYNC_LDS 1
#else
#define HAVE_ASYNC_LDS 0
#endif

#if HAVE_ASYNC_LDS
typedef int v4i __attribute__((vector_size(4 * sizeof(int))));
typedef __attribute__((address_space(3))) v4i* lds_v4i_ptr;
#endif

#if __has_builtin(__builtin_amdgcn_s_wait_asynccnt)
#define WAIT_ASYNC() __builtin_amdgcn_s_wait_asynccnt(0)
#else
#define WAIT_ASYNC() asm volatile("s_wait_asynccnt 0" ::: "memory")
#endif

#define W2T_STRIDE 36  // [p=144][o=32] padded: 16B-aligned b128 reads, 2-way store conflicts
#define W3T_STRIDE 20  // [p=288][o=16] padded:  8B-aligned b64  reads, 2-way store conflicts

__global__ __launch_bounds__(256) void addernet_fused(
    const float* __restrict__ x,    // [B,1,28,28]
    const float* __restrict__ w1,   // [16,9]
    const float* __restrict__ w2,   // [32,144]
    const float* __restrict__ w3,   // [16,288]
    const float* __restrict__ w4,   // [10,144]
    float* __restrict__ out)        // [B,10] log-softmax
{
    const int tid = threadIdx.x;
    const int b   = blockIdx.x;

    __shared__ __align__(16) float s_x[784];        // 28x28 raw input
    __shared__ float s_h1[16 * 15 * 15];            // L1 out, zero-padded 15x15
    __shared__ float s_h2[32 * 9 * 9];              // L2 out, zero-padded 9x9
    __shared__ float s_h3[16 * 4 * 4];              // L3 out, unpadded
    __shared__ __align__(16) float s_wt[288 * W3T_STRIDE]; // transposed w2 then w3
    __shared__ float s_log[10];
    __shared__ float s_lse;

    const float* xg = x + (size_t)b * 784;

    // ---- Stage 0 (all overlapped, one barrier):
    //  * async-copy x[b] into LDS (CDNA5 ASYNCcnt path)
    //  * zero padded activation buffers (padding contributes |0 - w|!)
    //  * repack w2 -> s_wt[p*36 + o] (transposed, padded stride)
#if HAVE_ASYNC_LDS
    if (tid < 196) {  // 196 lanes x 16B = 784 floats
        __builtin_amdgcn_global_load_async_to_lds_b128(
            (v4i*)(xg + tid * 4), (lds_v4i_ptr)(&s_x[tid * 4]), 0, 0);
    }
#else
    for (int k = tid; k < 784; k += 256) s_x[k] = xg[k];
#endif
    for (int k = tid; k < 16 * 15 * 15; k += 256) s_h1[k] = 0.0f;
    for (int k = tid; k < 32 * 9 * 9;  k += 256) s_h2[k] = 0.0f;
    for (int k = tid; k < 32 * 144; k += 256) {     // w2 transpose repack
        const int o = k / 144, p = k % 144;
        s_wt[p * W2T_STRIDE + o] = w2[k];
    }
#if HAVE_ASYNC_LDS
    WAIT_ASYNC();
#endif
    __syncthreads();

    // ---- Layer 1: 28x28 -> 16 x 13x13, k3 s2 p0, relu((.+10)/4).
    // The reference's (x - 0.5) is folded into the patch reads.
    if (tid < 169) {
        const int i = tid / 13, j = tid % 13;
        float v[9];
        const int base = (2 * i) * 28 + 2 * j;
#pragma unroll
        for (int ki = 0; ki < 3; ++ki)
#pragma unroll
            for (int kj = 0; kj < 3; ++kj)
                v[ki * 3 + kj] = s_x[base + ki * 28 + kj] - 0.5f;
        const int hb = (i + 1) * 15 + (j + 1);
        for (int o = 0; o < 16; ++o) {    // w1[o*9+k] uniform -> scalar loads
            float acc = 0.0f;
#pragma unroll
            for (int k = 0; k < 9; ++k) acc += fabsf(v[k] - w1[o * 9 + k]);
            const float h = (10.0f - acc) * 0.25f;
            s_h1[o * 225 + hb] = h > 0.0f ? h : 0.0f;
        }
    }
    __syncthreads();

    // ---- Layer 2 (dominant): 16x13x13 -> 32 x 7x7, k3 s2 p1, relu((.+130)/8)
    // 4 outputs x 4 locations per thread; weights via one ds_load_b128 per step.
    if (tid < 104) {
        const int og = tid & 7;   // o_base = og*4   (8 groups)
        const int lg = tid >> 3;  // l_base = lg*4   (13 groups, last partial)
        const int o_base = og * 4;
        float acc[4][4] = {};
        int lidx[4], lbase[4];
#pragma unroll
        for (int n = 0; n < 4; ++n) {
            int l = lg * 4 + n; if (l > 48) l = 48;   // duplicate tail, benign
            lidx[n]  = l;
            lbase[n] = 30 * (l / 7) + 2 * (l % 7);    // (2i)*15 + 2j in padded h1
        }
        for (int c = 0; c < 16; ++c) {
            const int cb = c * 225, pc = c * 9;
#pragma unroll
            for (int r = 0; r < 9; ++r) {
                const int ofs = (r / 3) * 15 + (r % 3);   // compile-time const
                float pv[4];
#pragma unroll
                for (int n = 0; n < 4; ++n) pv[n] = s_h1[cb + lbase[n] + ofs];
                const float4 wv = *reinterpret_cast<const float4*>(
                    &s_wt[(pc + r) * W2T_STRIDE + o_base]);
                const float wm[4] = {wv.x, wv.y, wv.z, wv.w};
#pragma unroll
                for (int m = 0; m < 4; ++m)
#pragma unroll
                    for (int n = 0; n < 4; ++n) acc[m][n] += fabsf(pv[n] - wm[m]);
            }
        }
#pragma unroll
        for (int m = 0; m < 4; ++m)
#pragma unroll
            for (int n = 0; n < 4; ++n) {
                const int l = lidx[n];
                const float h = (130.0f - acc[m][n]) * 0.125f;
                s_h2[(o_base + m) * 81 + (l / 7 + 1) * 9 + (l % 7 + 1)] =
                    h > 0.0f ? h : 0.0f;
            }
    }
    __syncthreads();

    // ---- Repack w3 -> s_wt[p*20 + o] (buffer reuse; L2 is done with it)
    for (int k = tid; k < 16 * 288; k += 256) {
        const int o = k / 288, p = k % 288;
        s_wt[p * W3T_STRIDE + o] = w3[k];
    }
    __syncthreads();

    // ---- Layer 3: 32x7x7 -> 16 x 4x4, k3 s2 p1, relu((.+280)/16)
    if (tid < 64) {
        const int og = tid & 7;   // o_base = og*2
        const int lg = tid >> 3;  // l_base = lg*2
        const int o_base = og * 2;
        float acc[2][2] = {};
        int lb[2];
#pragma unroll
        for (int n = 0; n < 2; ++n) {
            const int l = lg * 2 + n;
            lb[n] = 18 * (l >> 2) + 2 * (l & 3);      // (2i)*9 + 2j in padded h2
        }
        for (int c = 0; c < 32; ++c) {
            const int cb = c * 81, pc = c * 9;
#pragma unroll
            for (int r = 0; r < 9; ++r) {
                const int ofs = (r / 3) * 9 + (r % 3);
                float pv[2];
#pragma unroll
                for (int n = 0; n < 2; ++n) pv[n] = s_h2[cb + lb[n] + ofs];
                const float2 wv = *reinterpret_cast<const float2*>(
                    &s_wt[(pc + r) * W3T_STRIDE + o_base]);
                const float wm[2] = {wv.x, wv.y};
#pragma unroll
                for (int m = 0; m < 2; ++m)
#pragma unroll
                    for (int n = 0; n < 2; ++n) acc[m][n] += fabsf(pv[n] - wm[m]);
            }
        }
#pragma unroll
        for (int m = 0; m < 2; ++m)
#pragma unroll
            for (int n = 0; n < 2; ++n) {
                const float h = (280.0f - acc[m][n]) * 0.0625f;
                s_h3[(o_base + m) * 16 + (lg * 2 + n)] = h > 0.0f ? h : 0.0f;
            }
    }
    __syncthreads();

    // ---- Layer 4: 16x4x4 -> 10 logits (k3 s2 p0 -> single 3x3 window)
    if (tid < 10) {
        float acc = 0.0f;
        const float* __restrict__ w4p = w4 + tid * 144;
        for (int c = 0; c < 16; ++c) {
            const int cb = c * 16;
#pragma unroll
            for (int ki = 0; ki < 3; ++ki)
#pragma unroll
                for (int kj = 0; kj < 3; ++kj)
                    acc += fabsf(s_h3[cb + ki * 4 + kj] - w4p[c * 9 + ki * 3 + kj]);
        }
        s_log[tid] = -acc;
    }
    __syncthreads();

    // ---- log_softmax over 10 classes
    if (tid == 0) {
        float mx = s_log[0];
        for (int o = 1; o < 10; ++o) mx = fmaxf(mx, s_log[o]);
        float s = 0.0f;
        for (int o = 0; o < 10; ++o) s += expf(s_log[o] - mx);
        s_lse = mx + logf(s);
    }
    __syncthreads();
    if (tid < 10) out[(size_t)b * 10 + tid] = s_log[tid] - s_lse;
}

extern "C" void kernel_launch(void* const* d_in, const int* in_sizes, int n_in,
                              void* d_out, int out_size, void* d_ws, size_t ws_size,
                              hipStream_t stream) {
    const float* x  = (const float*)d_in[0];
    const float* w1 = (const float*)d_in[1];
    const float* w2 = (const float*)d_in[2];
    const float* w3 = (const float*)d_in[3];
    const float* w4 = (const float*)d_in[4];
    float* out = (float*)d_out;
    const int B = in_sizes[0] / 784;  // 4096
    addernet_fused<<<B, 256, 0, stream>>>(x, w1, w2, w3, w4, out);
    (void)n_in; (void)out_size; (void)d_ws; (void)ws_size;
}